// WHVIColumnMatrix_45861660786919
// MI455X (gfx1250) — compile-verified
//
#include <hip/hip_runtime.h>
#include <cmath>

// ---------------------------------------------------------------------------
// WHVI column-matrix kernel for MI455X (gfx1250).
//
// Reference reduces to:
//   g_tilde[k] = g_mu[k] + softplus(g_rho[k]) * eps[k]
//   u          = FWHT_8192(g_tilde)            (unnormalized Walsh-Hadamard)
//   w_col[j]   = s1[0] * s2[j] * u[j]
//   out[i,j]   = x[i] * w_col[j]               (4096 x 8192 f32, 128 MiB)
//
// Kernel 1 (1 workgroup, 512 thr = 16 waves): FWHT via
//   WHT_8192 = (H16 over i) o (H16 over j) o (H32 over k),  n = i*512 + j*32 + k
// H16 stages use V_WMMA_F32_16X16X4_F32 (exact f32, +-1 A-matrix built in regs).
// Kernel 2: pure streaming outer-product, b128 loads/stores (store-BW bound).
// ---------------------------------------------------------------------------

#define WHVI_D 8192
#define WHVI_B 4096

typedef __attribute__((ext_vector_type(2))) float v2f;
typedef __attribute__((ext_vector_type(8))) float v8f;

__global__ __launch_bounds__(512)
void whvi_fwht_prep(const float* __restrict__ s1,
                    const float* __restrict__ s2,
                    const float* __restrict__ g_mu,
                    const float* __restrict__ g_rho,
                    const float* __restrict__ eps,
                    float* __restrict__ w_col) {
    __shared__ float u[WHVI_D];             // 32 KB LDS
    const int tid = threadIdx.x;

    // --- g_tilde = g_mu + softplus(g_rho) * eps ---------------------------
    for (int n = tid; n < WHVI_D; n += 512) {
        float r  = g_rho[n];
        float sp = (r > 20.0f) ? r : log1pf(__expf(r));
        u[n] = g_mu[n] + sp * eps[n];
    }
    __syncthreads();

    // --- Stage K: 32-point WHT on each contiguous chunk (axis k) ----------
    if (tid < 256) {
        float v[32];
        const int base = tid * 32;
        #pragma unroll
        for (int q = 0; q < 32; ++q) v[q] = u[base + q];
        #pragma unroll
        for (int h = 1; h < 32; h <<= 1) {
            #pragma unroll
            for (int q = 0; q < 32; ++q) {
                if ((q & h) == 0) {
                    float a = v[q], b = v[q + h];
                    v[q] = a + b;
                    v[q + h] = a - b;
                }
            }
        }
        #pragma unroll
        for (int q = 0; q < 32; ++q) u[base + q] = v[q];
    }
    __syncthreads();

    // --- Stages J and I: Y = H16 * X (16 x 512) via WMMA f32 16x16x4 ------
    // stage 0 (axis j): addr(m, c) = (c>>5)*512 + m*32 + (c&31)
    // stage 1 (axis i): addr(m, c) = m*512 + c
    const int lane  = tid & 31;
    const int wave  = tid >> 5;
    const int laneM = lane & 15;
    const int khalf = (lane >> 4) << 1;     // 0 for lanes 0-15, 2 for 16-31
    const int mbase = (lane >> 4) << 3;     // D rows: M = mbase + r

    for (int stage = 0; stage < 2; ++stage) {
        #pragma unroll
        for (int s = 0; s < 2; ++s) {
            const int t = wave * 2 + s;         // tile 0..31 (16 columns each)
            const int c = t * 16 + laneM;       // this lane's column
            int cbase, cstride;                 // addr(m, c) = cbase + m*cstride
            if (stage == 0) { cbase = (c >> 5) * 512 + (c & 31); cstride = 32;  }
            else            { cbase = c;                          cstride = 512; }

            v8f acc = {};
            #pragma unroll
            for (int q = 0; q < 4; ++q) {       // chain K=4 WMMAs -> K=16
                const int k0 = 4 * q + khalf;
                v2f a, b;
                // A = H16: sign (-1)^popcount(m & k)
                a.x = (__popc(laneM & k0)       & 1) ? -1.0f : 1.0f;
                a.y = (__popc(laneM & (k0 + 1)) & 1) ? -1.0f : 1.0f;
                // B = X[k, c] from LDS
                b.x = u[cbase + k0 * cstride];
                b.y = u[cbase + (k0 + 1) * cstride];
                acc = __builtin_amdgcn_wmma_f32_16x16x4_f32(
                          false, a, false, b, (short)0, acc, false, false);
            }
            // D fragment: VGPR r -> row M = mbase + r, column N = laneM.
            // Columns are wave-exclusive, so in-place LDS update is safe.
            #pragma unroll
            for (int r = 0; r < 8; ++r)
                u[cbase + (mbase + r) * cstride] = acc[r];
        }
        __syncthreads();
    }

    // --- w_col = s1[0] * s2 * u ------------------------------------------
    const float s1_0 = s1[0];
    for (int n = tid; n < WHVI_D; n += 512)
        w_col[n] = s1_0 * s2[n] * u[n];
}

// Rank-1 outer product: out[i, :] = x[i] * w_col[:].  Pure store-bandwidth
// kernel: one b128 load of w (L2-resident, 32 KB), one b128 store per thread.
__global__ __launch_bounds__(256)
void whvi_outer(const float* __restrict__ x,
                const float* __restrict__ w,
                float4* __restrict__ out) {
    const size_t idx = (size_t)blockIdx.x * 256 + threadIdx.x;  // float4 index
    const int row = (int)(idx >> 11);      // D/4 = 2048 float4 per row
    const int col = (int)(idx & 2047);
    const float xv = x[row];
    const float4 wv = reinterpret_cast<const float4*>(w)[col];
    float4 o;
    o.x = xv * wv.x;
    o.y = xv * wv.y;
    o.z = xv * wv.z;
    o.w = xv * wv.w;
    out[idx] = o;
}

extern "C" void kernel_launch(void* const* d_in, const int* in_sizes, int n_in,
                              void* d_out, int out_size, void* d_ws, size_t ws_size,
                              hipStream_t stream) {
    const float* x     = (const float*)d_in[0];
    const float* s1    = (const float*)d_in[1];
    const float* s2    = (const float*)d_in[2];
    const float* g_mu  = (const float*)d_in[3];
    const float* g_rho = (const float*)d_in[4];
    const float* eps   = (const float*)d_in[5];
    float* out   = (float*)d_out;
    float* w_col = (float*)d_ws;            // 8192 floats of scratch

    whvi_fwht_prep<<<1, 512, 0, stream>>>(s1, s2, g_mu, g_rho, eps, w_col);

    const int total_f4 = (WHVI_B * WHVI_D) / 4;       // 8,388,608
    whvi_outer<<<total_f4 / 256, 256, 0, stream>>>(x, w_col, (float4*)out);
}